// multitask_MSE_ordinal_22643067584813
// MI455X (gfx1250) — compile-verified
//
#include <hip/hip_runtime.h>

// multitask ordinal MSE reduction for MI455X (gfx1250, wave32).
//
// Memory-bound: 192MB input + 12MB target streamed once -> ~8.8us floor at
// 23.3 TB/s. Main loop: b128 non-temporal coalesced loads, 8 VALU ops/elem.
// Block-level reduction uses V_WMMA_F32_16X16X4_F32 with a ones B-matrix
// (D[m,n] = sum_k A[m,k] + C) to row-sum 16 channels x 64 partial slots.
// Deterministic: no atomics; fixed-order final fold over block partials.

typedef __attribute__((ext_vector_type(2))) float v2f;
typedef __attribute__((ext_vector_type(4))) float v4f;
typedef __attribute__((ext_vector_type(8))) float v8f;

#define T_TASKS 3
#define C_DIM   16
#define NB_MAX  256
#define BLOCK   256

__global__ __launch_bounds__(BLOCK) void ordinal_mse_partial(
    const float* __restrict__ inp,   // [T, B, C] f32
    const int*   __restrict__ tgt,   // [B, T] int32 (harness ABI: integer -> int*)
    float*       __restrict__ ws,    // [T, NB, 2, C] block partials
    int Bn, int chunk, int NB)
{
    __shared__ float lds[2][C_DIM][65];   // [acc][channel][slot], padded vs bank conflicts

    const int t      = blockIdx.y;
    const int blk    = blockIdx.x;
    const int tid    = threadIdx.x;
    const int brow   = tid >> 2;          // 0..63 : b-offset within a 64-row tile
    const int c_base = (tid & 3) * 4;     // 0,4,8,12 : this lane's 4 channels

    float acc0[4] = {0.f, 0.f, 0.f, 0.f};
    float acc1[4] = {0.f, 0.f, 0.f, 0.f};

    const int b_start = blk * chunk;
    const int b_end   = min(Bn, b_start + chunk);
    const float* baseT = inp + (size_t)t * (size_t)Bn * C_DIM;

    for (int b = b_start + brow; b < b_end; b += 64) {
        // 512B contiguous per wave per iteration; stream (no reuse) -> non-temporal
        const v4f* p = (const v4f*)(baseT + (size_t)b * C_DIM + c_base);
        v4f x = __builtin_nontemporal_load(p);
        // prefetch ~32KB ahead of this block's stream
        __builtin_prefetch(baseT + (size_t)(b + 512) * C_DIM + c_base, 0, 0);

        const int t0 = tgt[(size_t)b * T_TASKS + 0];
        const int t1 = tgt[(size_t)b * T_TASKS + 1];
#pragma unroll
        for (int j = 0; j < 4; ++j) {
            const int c = c_base + j;
            const float m0 = (c <= t0) ? 1.0f : 0.0f;
            const float m1 = (c <= t1) ? 1.0f : 0.0f;
            const float d0 = x[j] - m0;
            const float d1 = x[j] - m1;
            acc0[j] = fmaf(d0, d0, acc0[j]);
            acc1[j] = fmaf(d1, d1, acc1[j]);
        }
    }

    // Stage per-lane partials: slot = brow (wave*8 + lane/4), 64 slots/channel.
#pragma unroll
    for (int j = 0; j < 4; ++j) {
        lds[0][c_base + j][brow] = acc0[j];
        lds[1][c_base + j][brow] = acc1[j];
    }
    __syncthreads();

    // Wave 0 reduces 64 slots/channel with the f32 matrix pipe.
    // A (16x4 f32) layout: lanes 0-15 hold M=lane, regs K=0,1; lanes 16-31 K=2,3.
    // B = ones(4x16) -> D[m,n] = sum_k A[m,k] + C[m,n]  (all columns identical).
    if (tid < 32) {
        const int m   = tid & 15;
        const int khi = tid >> 4;
        v8f s0 = {0.f, 0.f, 0.f, 0.f, 0.f, 0.f, 0.f, 0.f};
        v8f s1 = {0.f, 0.f, 0.f, 0.f, 0.f, 0.f, 0.f, 0.f};
        const v2f ones = {1.0f, 1.0f};
#pragma unroll
        for (int sb = 0; sb < 64; sb += 4) {
            const int s = sb + khi * 2;
            v2f a0 = { lds[0][m][s], lds[0][m][s + 1] };
            v2f a1 = { lds[1][m][s], lds[1][m][s + 1] };
            s0 = __builtin_amdgcn_wmma_f32_16x16x4_f32(
                     false, a0, false, ones, (short)0, s0, false, false);
            s1 = __builtin_amdgcn_wmma_f32_16x16x4_f32(
                     false, a1, false, ones, (short)0, s1, false, false);
        }
        // D 16x16 f32 layout: lane 0 reg r -> (M=r, N=0); lane 16 reg r -> (M=8+r, N=0)
        float* wsb = ws + ((size_t)(t * NB + blk) * 2) * C_DIM;
        if (tid == 0) {
#pragma unroll
            for (int r = 0; r < 8; ++r) {
                wsb[r]         = s0[r];
                wsb[C_DIM + r] = s1[r];
            }
        } else if (tid == 16) {
#pragma unroll
            for (int r = 0; r < 8; ++r) {
                wsb[8 + r]         = s0[r];
                wsb[C_DIM + 8 + r] = s1[r];
            }
        }
    }
}

__global__ __launch_bounds__(128) void ordinal_mse_finalize(
    const float* __restrict__ ws, float* __restrict__ out, int NB)
{
    __shared__ float l[2][T_TASKS][C_DIM];
    const int tid = threadIdx.x;
    if (tid < 2 * T_TASKS * C_DIM) {              // 96 threads: one (a,t,c) each
        const int a = tid / (T_TASKS * C_DIM);
        const int r = tid % (T_TASKS * C_DIM);
        const int t = r / C_DIM;
        const int c = r % C_DIM;
        float s = 0.f;
        for (int i = 0; i < NB; ++i)              // fixed order -> deterministic
            s += ws[((size_t)(t * NB + i) * 2 + a) * C_DIM + c];
        l[a][t][c] = s;
    }
    __syncthreads();
    if (tid < T_TASKS * C_DIM) {                  // 48 threads: one (t,c) each
        const int t = tid / C_DIM;
        const int c = tid % C_DIM;
        const float l0 = l[0][t][c];
        const float l1 = l[1][t][c];
        out[tid]                        = l0 + 2.0f * l1;  // loss = l0 + l1 + l2
        out[T_TASKS * C_DIM + tid]      = l0;              // l0
        out[2 * T_TASKS * C_DIM + tid]  = l1;              // l1
        out[3 * T_TASKS * C_DIM + tid]  = l1;              // l2 == l1
    }
}

extern "C" void kernel_launch(void* const* d_in, const int* in_sizes, int n_in,
                              void* d_out, int out_size, void* d_ws, size_t ws_size,
                              hipStream_t stream)
{
    const float* inp = (const float*)d_in[0];   // [T, B, C] f32
    const int*   tgt = (const int*)d_in[1];     // [B, T] int
    float*       out = (float*)d_out;           // 4 * T * C floats
    float*       ws  = (float*)d_ws;

    const int Bn = in_sizes[1] / T_TASKS;

    int NB = NB_MAX;
    while (NB > 1 &&
           (size_t)T_TASKS * (size_t)NB * 2 * C_DIM * sizeof(float) > ws_size)
        NB >>= 1;

    const int chunk = (Bn + NB - 1) / NB;

    dim3 grid(NB, T_TASKS);
    ordinal_mse_partial<<<grid, BLOCK, 0, stream>>>(inp, tgt, ws, Bn, chunk, NB);
    ordinal_mse_finalize<<<1, 128, 0, stream>>>(ws, out, NB);
}